// ChebConv_11776800325709
// MI455X (gfx1250) — compile-verified
//
#include <hip/hip_runtime.h>

#define V_NODES 50000
#define NNZ_E   800000
#define B_SZ    8
#define CIN     64
#define COUT    64
#define K_ORD   4
#define FDIM    (B_SZ * CIN)   // 512 = row width of each Chebyshev term

typedef float v2f __attribute__((ext_vector_type(2)));
typedef float v8f __attribute__((ext_vector_type(8)));

// ---------------------------------------------------------------------------
// 1) Transpose x (512, V) row-major -> x0 (V, 512) row-major, LDS tiled.
//    x[b,c,v] sits at r*V + v with r = b*64+c; x0[v, r] = x[r, v].
// ---------------------------------------------------------------------------
__global__ void __launch_bounds__(256)
transpose_kernel(const float* __restrict__ x, float* __restrict__ x0) {
    __shared__ float tile[32][33];
    const int v0 = blockIdx.x * 32;
    const int r0 = blockIdx.y * 32;
    const int tx = threadIdx.x;          // 0..31
    const int ty = threadIdx.y;          // 0..7
    #pragma unroll
    for (int i = 0; i < 32; i += 8) {
        int r = r0 + ty + i;
        int v = v0 + tx;
        float val = 0.0f;
        if (v < V_NODES) val = x[(size_t)r * V_NODES + v];   // coalesced in v
        tile[ty + i][tx] = val;
    }
    __syncthreads();
    #pragma unroll
    for (int i = 0; i < 32; i += 8) {
        int v = v0 + ty + i;
        int r = r0 + tx;
        if (v < V_NODES) x0[(size_t)v * FDIM + r] = tile[tx][ty + i]; // coalesced in r
    }
}

// ---------------------------------------------------------------------------
// 2) Zero-fill (float4 granularity)
// ---------------------------------------------------------------------------
__global__ void zero_kernel(float4* __restrict__ p, long n4) {
    long i = (long)blockIdx.x * blockDim.x + threadIdx.x;
    if (i < n4) p[i] = make_float4(0.f, 0.f, 0.f, 0.f);
}

// ---------------------------------------------------------------------------
// 3) SpMV scatter: y[rows[e], :] += vals[e] * x[cols[e], :]
//    One wave32 per edge; each lane handles 4 float4 chunks of the 512-wide
//    row. fp32 atomics (agent scope, relaxed) -> global_atomic_add_f32.
// ---------------------------------------------------------------------------
__global__ void __launch_bounds__(256)
spmv_scatter(const int* __restrict__ rows, const int* __restrict__ cols,
             const float* __restrict__ vals,
             const float* __restrict__ x, float* __restrict__ y) {
    int wave = (int)(((long)blockIdx.x * blockDim.x + threadIdx.x) >> 5);
    int lane = threadIdx.x & 31;
    if (wave >= NNZ_E) return;
    const int   r = rows[wave];
    const int   c = cols[wave];
    const float a = vals[wave];
    const float4* __restrict__ xr = (const float4*)(x + (size_t)c * FDIM);
    float* __restrict__ yr = y + (size_t)r * FDIM;
    #pragma unroll
    for (int j = 0; j < 4; ++j) {
        int idx = j * 32 + lane;            // float4 index 0..127, coalesced
        float4 xv = xr[idx];
        float* yp = yr + idx * 4;
        __hip_atomic_fetch_add(yp + 0, a * xv.x, __ATOMIC_RELAXED, __HIP_MEMORY_SCOPE_AGENT);
        __hip_atomic_fetch_add(yp + 1, a * xv.y, __ATOMIC_RELAXED, __HIP_MEMORY_SCOPE_AGENT);
        __hip_atomic_fetch_add(yp + 2, a * xv.z, __ATOMIC_RELAXED, __HIP_MEMORY_SCOPE_AGENT);
        __hip_atomic_fetch_add(yp + 3, a * xv.w, __ATOMIC_RELAXED, __HIP_MEMORY_SCOPE_AGENT);
    }
}

// ---------------------------------------------------------------------------
// 4) Chebyshev combine: y = 2*y - z   (y currently holds L @ x_prev)
// ---------------------------------------------------------------------------
__global__ void cheb_combine(float4* __restrict__ y, const float4* __restrict__ z, long n4) {
    long i = (long)blockIdx.x * blockDim.x + threadIdx.x;
    if (i < n4) {
        float4 a = y[i], b = z[i];
        y[i] = make_float4(2.f * a.x - b.x, 2.f * a.y - b.y,
                           2.f * a.z - b.z, 2.f * a.w - b.w);
    }
}

// ---------------------------------------------------------------------------
// 5) Dense stage via V_WMMA_F32_16X16X4_F32.
//    GEMM: M = V*B = 400000, Kdim = K_ORD*CIN = 256, N = COUT = 64.
//    A[m, k*64+c] = term_k[v*512 + b*64 + c], m = v*8+b.
//    One wave -> 16 rows x 64 cols (4 N-tiles). 64 k-steps of depth 4.
//    A frag (16x4 f32, 2 VGPRs): lanes 0-15 hold K=+0/+1, lanes 16-31 K=+2/+3.
//    B frag mirrored (4x16).
// ---------------------------------------------------------------------------
__global__ void __launch_bounds__(256)
cheb_gemm_wmma(const float* __restrict__ t0, const float* __restrict__ t1,
               const float* __restrict__ t2, const float* __restrict__ t3,
               const float* __restrict__ w,  const float* __restrict__ bias,
               float* __restrict__ out) {
    const int wave  = threadIdx.x >> 5;
    const int lane  = threadIdx.x & 31;
    const int mtile = blockIdx.x * 8 + wave;     // 0..24999 exactly
    const int mrow  = lane & 15;                 // M (for A) / N (for B,D)
    const int hi    = lane >> 4;                 // half-wave select

    const int m = mtile * 16 + mrow;
    const int v = m >> 3;
    const int b = m & 7;
    const size_t abase = (size_t)v * FDIM + b * 64;

    v8f acc[4] = {};                             // 4 N-tiles of 16

    #pragma unroll
    for (int k = 0; k < K_ORD; ++k) {
        const float* __restrict__ tk =
            (k == 0) ? t0 : (k == 1) ? t1 : (k == 2) ? t2 : t3;
        const float* __restrict__ wk = w + k * CIN * COUT;
        #pragma unroll
        for (int kc = 0; kc < 16; ++kc) {        // 16 depth-4 steps per term
            const int cbase = kc * 4 + hi * 2;
            v2f afrag;
            afrag.x = tk[abase + cbase + 0];
            afrag.y = tk[abase + cbase + 1];
            #pragma unroll
            for (int n = 0; n < 4; ++n) {
                const int ncol = n * 16 + mrow;
                v2f bfrag;
                bfrag.x = wk[(cbase + 0) * COUT + ncol];
                bfrag.y = wk[(cbase + 1) * COUT + ncol];
                acc[n] = __builtin_amdgcn_wmma_f32_16x16x4_f32(
                    /*neg_a=*/false, afrag, /*neg_b=*/false, bfrag,
                    /*c_mod=*/(short)0, acc[n],
                    /*reuse_a=*/false, /*reuse_b=*/false);
            }
        }
    }

    // D layout: VGPR r holds M = r + hi*8, N = n*16 + mrow. Output (B,COUT,V).
    #pragma unroll
    for (int n = 0; n < 4; ++n) {
        const int o = n * 16 + mrow;
        const float bo = bias[o];
        #pragma unroll
        for (int r2 = 0; r2 < 8; ++r2) {
            const int mm = mtile * 16 + r2 + hi * 8;
            const int vv = mm >> 3;
            const int bb = mm & 7;
            out[(size_t)bb * (COUT * (size_t)V_NODES) + (size_t)o * V_NODES + vv] =
                acc[n][r2] + bo;
        }
    }
}

// ---------------------------------------------------------------------------
extern "C" void kernel_launch(void* const* d_in, const int* in_sizes, int n_in,
                              void* d_out, int out_size, void* d_ws, size_t ws_size,
                              hipStream_t stream) {
    const float* x    = (const float*)d_in[0];
    const int*   rows = (const int*)  d_in[1];
    const int*   cols = (const int*)  d_in[2];
    const float* vals = (const float*)d_in[3];
    const float* w    = (const float*)d_in[4];
    const float* bias = (const float*)d_in[5];
    float*       out  = (float*)d_out;

    const size_t termElems = (size_t)V_NODES * FDIM;   // 25.6M floats = 102.4 MB
    float* x0 = (float*)d_ws;
    float* x1 = x0 + termElems;
    float* x2 = x1 + termElems;
    float* x3 = x2 + termElems;

    // 1) permute input into (V, 512)
    dim3 tb(32, 8);
    dim3 tg((V_NODES + 31) / 32, FDIM / 32);
    transpose_kernel<<<tg, tb, 0, stream>>>(x, x0);

    const long n4 = (long)(termElems / 4);
    const int  zb = 256;
    const long zg = (n4 + zb - 1) / zb;
    const int  sg = (int)(((long)NNZ_E * 32 + 255) / 256);   // 100000 blocks
    const long cg = (n4 + 255) / 256;

    // x1 = L x0
    zero_kernel<<<zg, zb, 0, stream>>>((float4*)x1, n4);
    spmv_scatter<<<sg, 256, 0, stream>>>(rows, cols, vals, x0, x1);

    // x2 = 2 (L x1) - x0
    zero_kernel<<<zg, zb, 0, stream>>>((float4*)x2, n4);
    spmv_scatter<<<sg, 256, 0, stream>>>(rows, cols, vals, x1, x2);
    cheb_combine<<<cg, 256, 0, stream>>>((float4*)x2, (const float4*)x0, n4);

    // x3 = 2 (L x2) - x1
    zero_kernel<<<zg, zb, 0, stream>>>((float4*)x3, n4);
    spmv_scatter<<<sg, 256, 0, stream>>>(rows, cols, vals, x2, x3);
    cheb_combine<<<cg, 256, 0, stream>>>((float4*)x3, (const float4*)x1, n4);

    // dense contraction + bias + output permute, fp32 WMMA
    cheb_gemm_wmma<<<3125, 256, 0, stream>>>(x0, x1, x2, x3, w, bias, out);
}